// WaveletGraph1DNonLinear_2216203125117
// MI455X (gfx1250) — compile-verified
//
#include <hip/hip_runtime.h>
#include <math.h>

// ---------------------------------------------------------------------------
// WaveletGraph1DNonLinear on MI455X (gfx1250, wave32, WMMA)
//
// out[b,i,d] = sum_j softmax_j(mask[i,j] ? scores[b,j,d] : -1e9) * feat[b,j,d]
//
// Restructured: with m[b,d] = max_j scores[b,j,d] (a valid softmax shift for
// every row i), E = exp(scores - m):
//     out = (mask @ (E*feat)) / (mask @ E)        -- two f32 mask-GEMMs
// executed with V_WMMA_F32_16X16X4_F32 (exact f32 path; mask is exactly 0/1).
// ---------------------------------------------------------------------------

typedef __attribute__((ext_vector_type(2))) float v2f;
typedef __attribute__((ext_vector_type(8))) float v8f;

#define BATCH 4
#define NODE  512
#define DIM   64

__device__ __forceinline__ float softplus_stable(float z) {
    // log(1 + exp(z)) without overflow
    return fmaxf(z, 0.0f) + log1pf(expf(-fabsf(z)));
}

// --- Kernel 1: m[b,d] = max over j of scores[b,j,d] ------------------------
// One wave (32 lanes) per (b,d) pair. 256 waves total.
__global__ void score_max_kernel(const float* __restrict__ x,
                                 const float* __restrict__ w0,
                                 const float* __restrict__ b0,
                                 const float* __restrict__ wa,
                                 const float* __restrict__ ba,
                                 float* __restrict__ mout) {
    const int bd   = blockIdx.x;       // 0..BATCH*DIM-1
    const int b    = bd >> 6;
    const int d    = bd & (DIM - 1);
    const int lane = threadIdx.x;      // 0..31 (wave32)

    const float c  = 0.70710678118654752440f * w0[0];
    const float bb = b0[0];
    const float w  = wa[d];
    const float ab = ba[d];

    const float* xb = x + (size_t)b * NODE * DIM + d;
    float mx = -INFINITY;
    for (int j = lane; j < NODE; j += 32) {
        const float f  = softplus_stable(fmaf(xb[(size_t)j * DIM], c, bb));
        const float sc = fmaf(f, w, ab);
        mx = fmaxf(mx, sc);
    }
    #pragma unroll
    for (int off = 16; off > 0; off >>= 1)
        mx = fmaxf(mx, __shfl_xor(mx, off, 32));
    if (lane == 0) mout[bd] = mx;
}

// --- Kernel 2: P = exp(scores - m) * feat, Q = exp(scores - m) -------------
__global__ void pq_kernel(const float* __restrict__ x,
                          const float* __restrict__ w0,
                          const float* __restrict__ b0,
                          const float* __restrict__ wa,
                          const float* __restrict__ ba,
                          const float* __restrict__ m,
                          float* __restrict__ P,
                          float* __restrict__ Q) {
    const int idx = blockIdx.x * blockDim.x + threadIdx.x;  // over b*NODE*DIM
    const int d   = idx & (DIM - 1);
    const int b   = idx >> 15;                              // / (NODE*DIM)

    const float f  = softplus_stable(fmaf(x[idx], 0.70710678118654752440f * w0[0], b0[0]));
    const float sc = fmaf(f, wa[d], ba[d]);
    const float e  = expf(sc - m[(b << 6) + d]);            // <= 1.0
    Q[idx] = e;
    P[idx] = e * f;
}

// --- Kernel 3: out = (mask @ P) / (mask @ Q) via v_wmma_f32_16x16x4_f32 ----
// One wave per 16x16 output tile: 4 batches * 32 i-tiles * 4 d-tiles = 512
// waves; 8 waves per 256-thread block -> 64 blocks. K loop: 512 in steps of 4,
// two independent accumulator chains (numerator / denominator).
//
// f32 A-matrix 16x4 layout (ISA 7.12.2): lane L (L&15 = row M);
//   lanes 0-15:  VGPR0 = K0,   VGPR1 = K1
//   lanes 16-31: VGPR0 = K2,   VGPR1 = K3
// B 4x16 uses the mirrored striping (lane = column N); C/D: VGPR r holds
// M = r + 8*(lane>=16), N = lane&15.
__global__ void attn_wmma_kernel(const int*   __restrict__ adj,
                                 const float* __restrict__ P,
                                 const float* __restrict__ Q,
                                 float*       __restrict__ out) {
    const int wave = (blockIdx.x * blockDim.x + threadIdx.x) >> 5;  // 0..511
    const int lane = threadIdx.x & 31;

    const int dtile = wave & 3;
    const int itile = (wave >> 2) & 31;
    const int b     = wave >> 7;
    const int i0    = itile << 4;
    const int d0    = dtile << 4;

    const int half = lane >> 4;         // 0: lanes 0-15, 1: lanes 16-31
    const int l15  = lane & 15;
    const int kOff = half << 1;         // K sub-offset {0,2} for this half

    const int*   arow = adj + (size_t)(i0 + l15) * NODE;            // A: row = M
    const float* Pb   = P + (size_t)b * NODE * DIM + d0 + l15;      // B: col = N
    const float* Qb   = Q + (size_t)b * NODE * DIM + d0 + l15;

    v8f cN = {};
    v8f cD = {};

    for (int k0 = 0; k0 < NODE; k0 += 4) {
        // A tile: mask[i0+l15, k0+kOff .. +1] as exact 0.0/1.0
        const int2 am = *(const int2*)(arow + k0 + kOff);   // 8B-aligned
        v2f a;
        a.x = (am.x > 0) ? 1.0f : 0.0f;
        a.y = (am.y > 0) ? 1.0f : 0.0f;

        const float* prow = Pb + (size_t)(k0 + kOff) * DIM;
        const float* qrow = Qb + (size_t)(k0 + kOff) * DIM;
        v2f bp, bq;
        bp.x = prow[0];   bp.y = prow[DIM];
        bq.x = qrow[0];   bq.y = qrow[DIM];

        // 8-arg pattern: (neg_a, A, neg_b, B, c_mod, C, reuse_a, reuse_b)
        cN = __builtin_amdgcn_wmma_f32_16x16x4_f32(false, a, false, bp,
                                                   (short)0, cN, false, false);
        cD = __builtin_amdgcn_wmma_f32_16x16x4_f32(false, a, false, bq,
                                                   (short)0, cD, false, false);
    }

    // out[b, i0+M, d0+N] = N/D   (D==0 only if a mask row is all-zero)
    float* orow = out + ((size_t)b * NODE + i0) * DIM + d0 + l15;
    #pragma unroll
    for (int r = 0; r < 8; ++r) {
        const float den = cD[r];
        const float val = (den != 0.0f) ? (cN[r] / den) : 0.0f;
        orow[(size_t)(r + (half << 3)) * DIM] = val;
    }
}

// ---------------------------------------------------------------------------
extern "C" void kernel_launch(void* const* d_in, const int* in_sizes, int n_in,
                              void* d_out, int out_size, void* d_ws, size_t ws_size,
                              hipStream_t stream) {
    (void)in_sizes; (void)n_in; (void)out_size; (void)ws_size;

    const float* x  = (const float*)d_in[0];
    const float* w0 = (const float*)d_in[1];
    const float* b0 = (const float*)d_in[2];
    const float* wa = (const float*)d_in[3];
    const float* ba = (const float*)d_in[4];
    const int*   adj = (const int*)d_in[5];
    float* out = (float*)d_out;

    float* P = (float*)d_ws;                     // BATCH*NODE*DIM floats
    float* Q = P + BATCH * NODE * DIM;           // BATCH*NODE*DIM floats
    float* m = Q + BATCH * NODE * DIM;           // BATCH*DIM floats

    score_max_kernel<<<BATCH * DIM, 32, 0, stream>>>(x, w0, b0, wa, ba, m);
    pq_kernel<<<(BATCH * NODE * DIM) / 256, 256, 0, stream>>>(x, w0, b0, wa, ba, m, P, Q);
    attn_wmma_kernel<<<64, 256, 0, stream>>>(adj, P, Q, out);
}